// CombineConcat_71365176590409
// MI455X (gfx1250) — compile-verified
//
#include <hip/hip_runtime.h>
#include <hip/hip_bf16.h>
#include <stdint.h>

typedef float fvec4 __attribute__((ext_vector_type(4)));
typedef int   iv4   __attribute__((ext_vector_type(4)));

#define NODES 128      // N
#define DIMF  256      // D floats
#define DF4   64       // float4 per D
#define ROWF4 128      // float4 per output row (2D)

#if __has_builtin(__builtin_amdgcn_global_load_async_to_lds_b128)
#define HAVE_ASYNC_LDS 1
#else
#define HAVE_ASYNC_LDS 0
#endif

__global__ __launch_bounds__(256) void combine_concat_kernel(
    const float* __restrict__ x1,
    const float* __restrict__ x2,
    float* __restrict__ out)
{
    const int tid = threadIdx.x;
    const int c   = tid & (ROWF4 - 1);   // float4 column within the 2D output row
    const int jh  = tid >> 7;            // which of the 2 rows this thread covers per step
    const int bi  = blockIdx.x;          // b*N + i
    const int b   = bi >> 7;

    __shared__ __align__(16) float x1row[DIMF];

    const float* x1p = x1 + (size_t)bi * DIMF;             // x1[b, i, :]
    const float* x2b = x2 + (size_t)b * NODES * DIMF;      // x2[b, :, :]
    float*       op  = out + (size_t)bi * NODES * (2 * DIMF);

    // ---- Stage the x1 row (1 KiB) into LDS via gfx1250 async global->LDS DMA ----
    if (tid < DF4) {
#if HAVE_ASYNC_LDS
        __builtin_amdgcn_global_load_async_to_lds_b128(
            (__attribute__((address_space(1))) iv4*)(x1p + (size_t)tid * 4),
            (__attribute__((address_space(3))) iv4*)(&x1row[tid * 4]),
            /*offset=*/0, /*cpol=*/0);
#else
        ((fvec4*)x1row)[tid] = ((const fvec4*)x1p)[tid];
#endif
    }
#if HAVE_ASYNC_LDS
#if __has_builtin(__builtin_amdgcn_s_wait_asynccnt)
    __builtin_amdgcn_s_wait_asynccnt(0);
#else
    asm volatile("s_wait_asynccnt 0" ::: "memory");
#endif
#endif
    __syncthreads();

    const bool isX1 = (c < DF4);         // uniform per wave32 (waves 0,1,4,5)
    fvec4 v1;
    if (isX1) v1 = ((const fvec4*)x1row)[c];   // one ds_load_b128, reused 64x

    const fvec4* x2f4 = (const fvec4*)x2b;
    fvec4*       of4  = (fvec4*)op;

    // 64 output rows per thread; x1-half waves: store-only; x2-half: L2 load + NT store.
    #pragma unroll 8
    for (int j = jh; j < NODES; j += 2) {
        fvec4 v;
        if (isX1) v = v1;
        else      v = x2f4[(size_t)j * DF4 + (c - DF4)];
        __builtin_nontemporal_store(v, &of4[(size_t)j * ROWF4 + c]);
    }
}

extern "C" void kernel_launch(void* const* d_in, const int* in_sizes, int n_in,
                              void* d_out, int out_size, void* d_ws, size_t ws_size,
                              hipStream_t stream) {
    const float* x1 = (const float*)d_in[0];
    const float* x2 = (const float*)d_in[1];
    float* out = (float*)d_out;

    // in_sizes[0] = B*N*D ; blocks = B*N (one (b,i) pair per block)
    const int n_bi = in_sizes[0] / DIMF;
    combine_concat_kernel<<<n_bi, 256, 0, stream>>>(x1, x2, out);
}